// SlimTikNetworkMNIST_34617436406083
// MI455X (gfx1250) — compile-verified
//
#include <hip/hip_runtime.h>

typedef __attribute__((ext_vector_type(2))) float v2f;
typedef __attribute__((ext_vector_type(8))) float v8f;

#define LAMBDA 0.05f

// D = A(16x4) * B(4x16) + C, all fp32 (V_WMMA_F32_16X16X4_F32)
static __device__ __forceinline__ v8f wmma_f32_k4(v2f a, v2f b, v8f c) {
    return __builtin_amdgcn_wmma_f32_16x16x4_f32(
        /*neg_a=*/false, a, /*neg_b=*/false, b,
        /*c_mod=*/(short)0, c, /*reuse_a=*/false, /*reuse_b=*/false);
}

// padded-store offsets (used only in the table-init kernel)
static __device__ __forceinline__ int offA(int r) {   // 14x14 -> 16x16 pad, 4096/img
    const int n = r / 196, pp = r % 196;
    return n * 4096 + (pp / 14 + 1) * 16 + (pp % 14) + 1;
}
static __device__ __forceinline__ int offB(int r) {   // 7x7 -> 9x9 pad, 324/img
    const int n = r / 49, pp = r % 49;
    return n * 324 + (pp / 7 + 1) * 9 + (pp % 7) + 1;
}

// ---------------------------------------------------------------------------
// Workspace layout (floats):
//   xpad    @ 0        : 256*900            (N,30,30)    pad=1
//   h1pad   @ 230400   : 256*16*256         (N,16,16,16) pad=1
//   h2pad   @ 1278976  : 256*4*81           (N,4,9,9)    pad=1
//   featpad @ 1361920  : 256*16*256         (N,16,16,16) pad=1
//   rpad    @ 2410496  : 256*900            (N,30,30)    pad=1
//   slack   @ 2640896  : 64                 (zeroed overrun pad)
//   w2pad   @ 2640960  : 16*256             (oc-padded w2; fully written)
//   tabs    @ 2645056  : 1568 ints          (offA[784], offB[784]; fully written)
// Zeroed each launch: [230400, 2640960)
// ---------------------------------------------------------------------------
#define OFF_XPAD    0
#define OFF_H1PAD   230400
#define OFF_H2PAD   1278976
#define OFF_FEATPAD 1361920
#define OFF_RPAD    2410496
#define OFF_W2PAD   2640960
#define OFF_TABS    2645056
#define ZERO_CNT4   602640   // (2640960-230400)/4

__global__ __launch_bounds__(256) void k_zero(float4* __restrict__ p) {
    const int i = blockIdx.x * 256 + threadIdx.x;
    if (i < ZERO_CNT4) p[i] = make_float4(0.f, 0.f, 0.f, 0.f);
}

__global__ void k_init_grad(const float* __restrict__ Wf,
                            const float* __restrict__ fb,
                            float* __restrict__ grad) {
    int i = blockIdx.x * blockDim.x + threadIdx.x;
    if (i < 256)       grad[i]   = LAMBDA * Wf[i];
    else if (i == 256) grad[256] = LAMBDA * fb[0];
}

// zero-pad x: (N,1,28,28) -> (N,30,30)
__global__ __launch_bounds__(256) void k_pad_x(const float* __restrict__ x,
                                               float* __restrict__ xp) {
    const int idx = blockIdx.x * 256 + threadIdx.x;    // 230400
    const int n = idx / 900, rr = idx % 900;
    const int ih = rr / 30, iw = rr % 30;
    float v = 0.f;
    if (ih >= 1 && ih <= 28 && iw >= 1 && iw <= 28)
        v = x[n * 784 + (ih - 1) * 28 + (iw - 1)];
    xp[idx] = v;
}

// pad w2 (4,16,4,4) -> (16,256) and build store-offset tables
__global__ __launch_bounds__(256) void k_prep(const float* __restrict__ w2,
                                              float* __restrict__ w2p,
                                              int* __restrict__ tab) {
    const int i = blockIdx.x * 256 + threadIdx.x;      // 0..6143 (24 blocks)
    if (i < 4096) {
        const int co = i >> 8;
        w2p[i] = (co < 4) ? w2[i] : 0.f;
    } else if (i < 4880) {
        tab[i - 4096] = offA(i - 4096);                // tabA[0..783]
    } else if (i < 5664) {
        tab[i - 4096] = offB(i - 4880);                // tabB[784..1567]
    }
}

// ---------------------------------------------------------------------------
// conv1: xpad -> relu -> h1pad interior. GEMM M=16 pixels, N=16 oc, K=16.
// 4 images/group -> 49 exact tiles; epilogue offsets via b128 table loads.
// ---------------------------------------------------------------------------
__global__ __launch_bounds__(32) void k_conv1(const float* __restrict__ xp,
                                              const float* __restrict__ w1,
                                              const float* __restrict__ b1,
                                              float* __restrict__ h1p,
                                              const int* __restrict__ tabA) {
    const int lane  = threadIdx.x;
    const int half  = lane >> 4;
    const int kbase = half << 1;
    const int Ml    = lane & 15;
    const int g   = blockIdx.x / 49;          // scalar, 4 images
    const int t49 = blockIdx.x % 49;          // scalar

    const int prow = t49 * 16 + Ml;           // 0..783
    const int n_in = prow / 196, p = prow % 196;
    const int ho = p / 14, wo = p % 14;

    const float* xin = xp + g * 3600 + n_in * 900 + (2 * ho) * 30 + 2 * wo + kbase;
    const float* wbp = w1 + Ml * 16 + kbase;

    v8f acc = {};
#pragma unroll
    for (int s = 0; s < 4; ++s) {
        const v2f a = *(const v2f*)(xin + s * 30);
        const v2f b = *(const v2f*)(wbp + s * 4);
        acc = wmma_f32_k4(a, b, acc);
    }

    // two aligned b128 loads give this lane's 8 store offsets
    const int  rbase = t49 * 16 + (half << 3);
    const int4 t0 = *(const int4*)(tabA + rbase);
    const int4 t1 = *(const int4*)(tabA + rbase + 4);
    const int  offv[8] = { t0.x, t0.y, t0.z, t0.w, t1.x, t1.y, t1.z, t1.w };

    const float bias  = b1[Ml];
    const int   lbase = Ml * 256;             // co = Ml
    float* obase = h1p + g * 16384;           // scalar base -> saddr stores
#pragma unroll
    for (int v = 0; v < 8; ++v) {
        float val = acc[v] + bias;
        obase[lbase + offv[v]] = val > 0.f ? val : 0.f;
    }
}

// ---------------------------------------------------------------------------
// conv2: h1pad -> relu -> h2pad interior. GEMM M=16 pixels, N=16 (oc padded
// 4->16), K=256. 16 images/group -> 49 exact tiles; immediate-offset b64 loads.
// ---------------------------------------------------------------------------
__global__ __launch_bounds__(32) void k_conv2(const float* __restrict__ h1p,
                                              const float* __restrict__ w2p,
                                              const float* __restrict__ b2,
                                              float* __restrict__ h2p,
                                              const int* __restrict__ tabB) {
    const int lane  = threadIdx.x;
    const int half  = lane >> 4;
    const int kbase = half << 1;
    const int Ml    = lane & 15;
    const int g   = blockIdx.x / 49;          // scalar, 16 images
    const int t49 = blockIdx.x % 49;          // scalar

    const int prow = t49 * 16 + Ml;           // 0..783
    const int n_in = prow / 49, p = prow % 49;
    const int ho = p / 7, wo = p % 7;

    const float* hin = h1p + g * 65536 + n_in * 4096 + (2 * ho) * 16 + 2 * wo + kbase;
    const float* wbp = w2p + Ml * 256 + kbase;

    v8f acc = {};
#pragma unroll
    for (int s = 0; s < 64; ++s) {            // k = s*4+kbase: ci=s>>2, kh=s&3, kw=kbase(+1)
        const v2f a = *(const v2f*)(hin + (s >> 2) * 256 + (s & 3) * 16);
        const v2f b = *(const v2f*)(wbp + s * 4);
        acc = wmma_f32_k4(a, b, acc);
    }

    if (Ml < 4) {                             // only 4 real output channels
        const int  rbase = t49 * 16 + (half << 3);
        const int4 t0 = *(const int4*)(tabB + rbase);
        const int4 t1 = *(const int4*)(tabB + rbase + 4);
        const int  offv[8] = { t0.x, t0.y, t0.z, t0.w, t1.x, t1.y, t1.z, t1.w };

        const float bias  = b2[Ml];
        const int   lbase = Ml * 81;
        float* obase = h2p + g * 5184;
#pragma unroll
        for (int v = 0; v < 8; ++v) {
            float val = acc[v] + bias;
            obase[lbase + offv[v]] = val > 0.f ? val : 0.f;
        }
    }
}

// ---------------------------------------------------------------------------
// deconv: h2pad -> convT(4->16) -> relu -> featpad interior.
// GEMM M=16 pixels, N=16 oc, K=64. Parity -> cndmask; immediate ci*81 offsets.
// ---------------------------------------------------------------------------
__global__ __launch_bounds__(32) void k_deconv(const float* __restrict__ h2p,
                                               const float* __restrict__ wd,
                                               const float* __restrict__ bd,
                                               float* __restrict__ featp,
                                               const int* __restrict__ tabA) {
    const int lane  = threadIdx.x;
    const int half  = lane >> 4;
    const int kbase = half << 1;
    const int Ml    = lane & 15;
    const int g   = blockIdx.x / 49;          // scalar, 4 images
    const int t49 = blockIdx.x % 49;          // scalar

    const int prow = t49 * 16 + Ml;           // 0..783
    const int n_in = prow / 196, p = prow % 196;
    const int ho = p / 14, wo = p % 14;
    const float* hgrp = h2p + g * 1296;       // scalar base
    const int nboff = n_in * 324;             // per-lane 32-bit

    int  coloff[2]; bool wok[2];
#pragma unroll
    for (int j = 0; j < 2; ++j) {
        const int ww = wo + 1 - (kbase + j);
        wok[j]    = ((ww & 1) == 0);
        coloff[j] = (ww >> 1) + 1;            // in [0,8]
    }
    int  rowoff[4]; bool hok[4];
#pragma unroll
    for (int kh = 0; kh < 4; ++kh) {
        const int hh = ho + 1 - kh;
        hok[kh]    = ((hh & 1) == 0);
        rowoff[kh] = ((hh >> 1) + 1) * 9;     // row in [0,8]
    }
    const float* wbp = wd + Ml * 16 + kbase;

    v8f acc = {};
#pragma unroll
    for (int s = 0; s < 16; ++s) {
        const int ci = s >> 2, kh = s & 3;
        v2f a, b;
#pragma unroll
        for (int j = 0; j < 2; ++j) {
            const float v = hgrp[nboff + ci * 81 + rowoff[kh] + coloff[j]];
            a[j] = (hok[kh] && wok[j]) ? v : 0.f;
        }
        b = *(const v2f*)(wbp + ci * 256 + kh * 4);
        acc = wmma_f32_k4(a, b, acc);
    }

    const int  rbase = t49 * 16 + (half << 3);
    const int4 t0 = *(const int4*)(tabA + rbase);
    const int4 t1 = *(const int4*)(tabA + rbase + 4);
    const int  offv[8] = { t0.x, t0.y, t0.z, t0.w, t1.x, t1.y, t1.z, t1.w };

    const float bias  = bd[Ml];
    const int   lbase = Ml * 256;
    float* obase = featp + g * 16384;
#pragma unroll
    for (int v = 0; v < 8; ++v) {
        float val = acc[v] + bias;
        obase[lbase + offv[v]] = val > 0.f ? val : 0.f;
    }
}

// ---------------------------------------------------------------------------
// out = convT(feat, W, fb) == z@wb ; r = out - c -> rpad interior;
// block-reduce sum(r) -> grad[256]. Parity hoisted: exactly 2x2 taps, no ifs.
// ---------------------------------------------------------------------------
__global__ __launch_bounds__(256) void k_out_r(const float* __restrict__ featp,
                                               const float* __restrict__ Wf,
                                               const float* __restrict__ fb,
                                               const float* __restrict__ cin,
                                               float* __restrict__ out,
                                               float* __restrict__ rp,
                                               float* __restrict__ grad) {
    __shared__ float sdata[256];
    const int tid = blockIdx.x * 256 + threadIdx.x;   // 200704
    const int n = tid / 784, p = tid % 784;
    const int ho = p / 28, wo = p % 28;
    const int kh0 = (ho + 1) & 1;
    const int kw0 = (wo + 1) & 1;
    const float* fn = featp + n * 4096;
    float sum = fb[0];
    for (int c = 0; c < 16; ++c) {
        const float* fc = fn + c * 256;
        const float* wc = Wf + c * 16;
#pragma unroll
        for (int th = 0; th < 2; ++th) {
            const int kh = kh0 + 2 * th;
            const int hi = (ho + 1 - kh) >> 1;        // +1 -> [0,15]
#pragma unroll
            for (int tw = 0; tw < 2; ++tw) {
                const int kw = kw0 + 2 * tw;
                const int wi = (wo + 1 - kw) >> 1;
                sum += fc[(hi + 1) * 16 + (wi + 1)] * wc[kh * 4 + kw];
            }
        }
    }
    out[tid] = sum;
    const float rv = sum - cin[tid];
    rp[n * 900 + (ho + 1) * 30 + (wo + 1)] = rv;

    sdata[threadIdx.x] = rv;
    __syncthreads();
    for (int s2 = 128; s2 > 0; s2 >>= 1) {
        if (threadIdx.x < s2) sdata[threadIdx.x] += sdata[threadIdx.x + s2];
        __syncthreads();
    }
    if (threadIdx.x == 0) atomicAdd(grad + 256, sdata[0]);
}

// ---------------------------------------------------------------------------
// wb_grad[0..255]: G[c][kpos] = sum_rows feat * r  (transpose-conv dW).
// GEMM M=16 channels, N=16 kpos. K enumerated as p' = hi*16 + wi' (wi' 0..15,
// cols 14/15 hit featpad zero border -> contribute 0). A address is linear in
// p'; full unroll folds all addresses into immediates. One wave per image.
// ---------------------------------------------------------------------------
__global__ __launch_bounds__(32) void k_wbgrad(const float* __restrict__ featp,
                                               const float* __restrict__ rp,
                                               float* __restrict__ grad) {
    const int lane  = threadIdx.x;
    const int half  = lane >> 4;
    const int kbase = half << 1;
    const int Ml    = lane & 15;          // A: channel / B: kernel position
    const int n     = blockIdx.x;         // 0..255
    const float* fnp = featp + n * 4096 + Ml * 256 + 17;  // +17: linear in p'
    const float* rn  = rp + n * 900 + (Ml >> 2) * 30 + (Ml & 3);

    v8f acc = {};
#pragma unroll
    for (int s = 0; s < 56; ++s) {
        const int p0 = s * 4 + kbase;                 // even
        const int v0 = (p0 >> 4) * 60 + (p0 & 15) * 2;
        v2f a, b;
        a[0] = fnp[p0];
        a[1] = fnp[p0 + 1];
        b[0] = rn[v0];
        b[1] = rn[v0 + 2];
        acc = wmma_f32_k4(a, b, acc);
    }
#pragma unroll
    for (int v = 0; v < 8; ++v) {
        const int c = v + (half << 3);
        atomicAdd(grad + c * 16 + Ml, acc[v]);
    }
}

// ---------------------------------------------------------------------------
extern "C" void kernel_launch(void* const* d_in, const int* in_sizes, int n_in,
                              void* d_out, int out_size, void* d_ws, size_t ws_size,
                              hipStream_t stream) {
    const float* x  = (const float*)d_in[0];
    const float* c  = (const float*)d_in[1];
    const float* w1 = (const float*)d_in[2];
    const float* b1 = (const float*)d_in[3];
    const float* w2 = (const float*)d_in[4];
    const float* b2 = (const float*)d_in[5];
    const float* wd = (const float*)d_in[6];
    const float* bd = (const float*)d_in[7];
    const float* W  = (const float*)d_in[8];
    const float* fb = (const float*)d_in[9];

    float* out  = (float*)d_out;           // 200704 floats
    float* grad = out + 200704;            // 257 floats (wb_grad)

    float* ws    = (float*)d_ws;
    float* xpad  = ws + OFF_XPAD;
    float* h1p   = ws + OFF_H1PAD;
    float* h2p   = ws + OFF_H2PAD;
    float* featp = ws + OFF_FEATPAD;
    float* rpad  = ws + OFF_RPAD;
    float* w2pad = ws + OFF_W2PAD;
    int*   tabs  = (int*)(ws + OFF_TABS);
    const int* tabA = tabs;
    const int* tabB = tabs + 784;

    k_zero     <<<(ZERO_CNT4 + 255) / 256, 256, 0, stream>>>((float4*)h1p);
    k_init_grad<<<1, 288, 0, stream>>>(W, fb, grad);
    k_pad_x    <<<900, 256, 0, stream>>>(x, xpad);
    k_prep     <<<24, 256, 0, stream>>>(w2, w2pad, tabs);
    k_conv1    <<<3136, 32, 0, stream>>>(xpad, w1, b1, h1p, tabA);
    k_conv2    <<<784, 32, 0, stream>>>(h1p, w2pad, b2, h2p, tabB);
    k_deconv   <<<3136, 32, 0, stream>>>(h2p, wd, bd, featp, tabA);
    k_out_r    <<<784, 256, 0, stream>>>(featp, W, fb, c, out, rpad, grad);
    k_wbgrad   <<<256, 32, 0, stream>>>(featp, rpad, grad);
}